// LargeLoss_29566554866291
// MI455X (gfx1250) — compile-verified
//
#include <hip/hip_runtime.h>
#include <math.h>

// ---------------------------------------------------------------------------
// LargeLoss (ASL-style hard-thresholding loss), MI455X / gfx1250.
//
// Bandwidth-bound elementwise + exact k-th-largest radix select.
// No matmul structure => WMMA inapplicable; the MI455X-rewarded path is
// wave32 + global_load_b128 streaming + LDS ds_add histogram atomics +
// global_prefetch, with the selection scan kept LDS-resident (latency).
// ---------------------------------------------------------------------------

namespace {

constexpr int BINS_P01 = 2048;  // 11-bit digits, passes 0 and 1
constexpr int BINS_P2  = 1024;  // 10-bit digit, pass 2
constexpr int NHIST    = BINS_P01 + BINS_P01 + BINS_P2;

struct SelState {
    unsigned prefix;   // radix prefix selected so far
    unsigned kk;       // remaining rank within current prefix
    unsigned active;   // k > 0 ?
    float    thr;      // final threshold (k-th largest unobserved loss)
};

// One transcendental pair serves BOTH losses:
//   bce( x, t) = max( x,0) - x*t + log1p(exp(-|x|))
//   bce(-x,-t) = max(-x,0) - x*t + log1p(exp(-|x|))
__device__ __forceinline__ void bce_pair(float x, float t, float& pl, float& nl) {
    const float s  = log1pf(expf(-fabsf(x)));
    const float xt = x * t;
    pl = fmaxf(x, 0.0f) - xt + s;
    nl = fmaxf(-x, 0.0f) - xt + s;
}

// unobserved_loss = negative_mask * bce(-x,-t); bit-exact vs 1.0f*nl / 0.0f.
// Always >= +0.0f, so uint bit pattern orders identically to float value.
__device__ __forceinline__ unsigned unobs_key(float x, float t) {
    float pl, nl;
    bce_pair(x, t, pl, nl);
    return (t < 0.0f) ? __float_as_uint(nl) : 0u;
}

template <int PASS>
__device__ __forceinline__ bool digit(unsigned key, unsigned prefix, unsigned& bin) {
    if (PASS == 0)      { bin = key >> 21;           return true; }
    else if (PASS == 1) { bin = (key >> 10) & 0x7FFu; return (key >> 21) == prefix; }
    else                { bin = key & 0x3FFu;         return (key >> 10) == prefix; }
}

// ---------------------------------------------------------------------------
__global__ void init_kernel(unsigned* __restrict__ hists,
                            SelState* __restrict__ st,
                            const int* __restrict__ kptr) {
    int i = blockIdx.x * blockDim.x + threadIdx.x;
    if (i < NHIST) hists[i] = 0u;
    if (i == 0) {
        int k = *kptr;
        st->prefix = 0u;
        st->kk     = (unsigned)(k > 0 ? k : 0);
        st->active = (k > 0) ? 1u : 0u;
        st->thr    = __uint_as_float(0x7F800000u);  // +inf: k==0 => mask all-pass
    }
}

// ---------------------------------------------------------------------------
// Pass 0: compute keys, histogram top 11 bits; optionally cache keys in ws.
template <bool STORE>
__global__ void pass0_kernel(const float4* __restrict__ xin,
                             const float4* __restrict__ tin,
                             int n4, int n,
                             unsigned* __restrict__ keys,   // may be null if !STORE
                             unsigned* __restrict__ ghist,
                             const SelState* __restrict__ st) {
    __shared__ unsigned lh[BINS_P01];
    for (int i = threadIdx.x; i < BINS_P01; i += blockDim.x) lh[i] = 0u;
    __syncthreads();

    if (st->active) {
        const int stride = gridDim.x * blockDim.x;
        const int gid = blockIdx.x * blockDim.x + threadIdx.x;
        for (int i = gid; i < n4; i += stride) {
            if (i + stride < n4) {
                __builtin_prefetch(xin + i + stride, 0, 0);  // global_prefetch_b8
                __builtin_prefetch(tin + i + stride, 0, 0);
            }
            float4 x = xin[i];
            float4 t = tin[i];
            uint4 kv;
#pragma unroll
            for (int c = 0; c < 4; ++c) {
                const unsigned key = unobs_key((&x.x)[c], (&t.x)[c]);
                (&kv.x)[c] = key;
                atomicAdd(&lh[key >> 21], 1u);  // ds_add_u32
            }
            if (STORE) ((uint4*)keys)[i] = kv;  // global_store_b128
        }
        // scalar tail
        int i = n4 * 4 + gid;
        if (i < n) {
            const float* xs = (const float*)xin;
            const float* ts = (const float*)tin;
            const unsigned key = unobs_key(xs[i], ts[i]);
            if (STORE) keys[i] = key;
            atomicAdd(&lh[key >> 21], 1u);
        }
    }
    __syncthreads();
    for (int i = threadIdx.x; i < BINS_P01; i += blockDim.x) {
        unsigned v = lh[i];
        if (v) atomicAdd(&ghist[i], v);
    }
}

// ---------------------------------------------------------------------------
// Passes 1/2, cached keys: stream 32-bit keys (half the bytes, no VALU math).
template <int PASS>
__global__ void hist_cached(const uint4* __restrict__ keys4,
                            int n4, int n,
                            unsigned* __restrict__ ghist,
                            const SelState* __restrict__ st) {
    constexpr int NB = (PASS == 2) ? BINS_P2 : BINS_P01;
    __shared__ unsigned lh[NB];
    for (int i = threadIdx.x; i < NB; i += blockDim.x) lh[i] = 0u;
    __syncthreads();

    if (st->active) {
        const unsigned prefix = st->prefix;
        const int stride = gridDim.x * blockDim.x;
        const int gid = blockIdx.x * blockDim.x + threadIdx.x;
        for (int i = gid; i < n4; i += stride) {
            if (i + stride < n4) __builtin_prefetch(keys4 + i + stride, 0, 0);
            uint4 kv = keys4[i];
#pragma unroll
            for (int c = 0; c < 4; ++c) {
                unsigned bin;
                if (digit<PASS>((&kv.x)[c], prefix, bin)) atomicAdd(&lh[bin], 1u);
            }
        }
        int i = n4 * 4 + gid;
        if (i < n) {
            unsigned bin;
            if (digit<PASS>(((const unsigned*)keys4)[i], prefix, bin))
                atomicAdd(&lh[bin], 1u);
        }
    }
    __syncthreads();
    for (int i = threadIdx.x; i < NB; i += blockDim.x) {
        unsigned v = lh[i];
        if (v) atomicAdd(&ghist[i], v);
    }
}

// ---------------------------------------------------------------------------
// Passes 1/2, fallback: recompute keys from x,t (small workspace case).
template <int PASS>
__global__ void hist_recompute(const float4* __restrict__ xin,
                               const float4* __restrict__ tin,
                               int n4, int n,
                               unsigned* __restrict__ ghist,
                               const SelState* __restrict__ st) {
    constexpr int NB = (PASS == 2) ? BINS_P2 : BINS_P01;
    __shared__ unsigned lh[NB];
    for (int i = threadIdx.x; i < NB; i += blockDim.x) lh[i] = 0u;
    __syncthreads();

    if (st->active) {
        const unsigned prefix = st->prefix;
        const int stride = gridDim.x * blockDim.x;
        const int gid = blockIdx.x * blockDim.x + threadIdx.x;
        for (int i = gid; i < n4; i += stride) {
            if (i + stride < n4) {
                __builtin_prefetch(xin + i + stride, 0, 0);
                __builtin_prefetch(tin + i + stride, 0, 0);
            }
            float4 x = xin[i];
            float4 t = tin[i];
#pragma unroll
            for (int c = 0; c < 4; ++c) {
                unsigned bin;
                if (digit<PASS>(unobs_key((&x.x)[c], (&t.x)[c]), prefix, bin))
                    atomicAdd(&lh[bin], 1u);
            }
        }
        int i = n4 * 4 + gid;
        if (i < n) {
            unsigned bin;
            if (digit<PASS>(unobs_key(((const float*)xin)[i], ((const float*)tin)[i]),
                            prefix, bin))
                atomicAdd(&lh[bin], 1u);
        }
    }
    __syncthreads();
    for (int i = threadIdx.x; i < NB; i += blockDim.x) {
        unsigned v = lh[i];
        if (v) atomicAdd(&ghist[i], v);
    }
}

// ---------------------------------------------------------------------------
// LDS-resident top-down select: parallel load + chunk sums, then a short
// serial walk over <=256 chunk sums + <=8 bins. Avoids the dependent-global-
// load latency chain a naive serial scan would incur.
template <int PASS>
__global__ void scan_pass(const unsigned* __restrict__ ghist,
                          SelState* __restrict__ st) {
    constexpr int NB   = (PASS == 2) ? BINS_P2 : BINS_P01;
    constexpr int BITS = (PASS == 2) ? 10 : 11;
    constexpr int T    = 256;
    constexpr int PER  = NB / T;

    if (!st->active) return;  // uniform

    __shared__ unsigned h[NB];
    __shared__ unsigned csum[T];

    const int t = threadIdx.x;
    unsigned s = 0;
#pragma unroll
    for (int j = 0; j < PER; ++j) {
        unsigned v = ghist[t * PER + j];
        h[t * PER + j] = v;
        s += v;
    }
    csum[t] = s;
    __syncthreads();

    if (t == 0) {
        const unsigned kk = st->kk;
        unsigned cum = 0;
        int selc = 0;
        for (int c = T - 1; c >= 0; --c) {
            unsigned v = csum[c];
            if (cum + v >= kk) { selc = c; break; }
            cum += v;
        }
        int sel = selc * PER;
        for (int b = selc * PER + PER - 1; b >= selc * PER; --b) {
            unsigned v = h[b];
            if (cum + v >= kk) { sel = b; break; }
            cum += v;
        }
        st->prefix = (st->prefix << BITS) | (unsigned)sel;
        st->kk = kk - cum;
        if (PASS == 2) st->thr = __uint_as_float(st->prefix);  // exact k-th largest
    }
}

// ---------------------------------------------------------------------------
__global__ void final_loss(const float4* __restrict__ xin,
                           const float4* __restrict__ tin,
                           float4* __restrict__ out,
                           int n4, int n,
                           const SelState* __restrict__ st) {
    const float thr = st->thr;
    const int stride = gridDim.x * blockDim.x;
    const int gid = blockIdx.x * blockDim.x + threadIdx.x;

    for (int i = gid; i < n4; i += stride) {
        if (i + stride < n4) {
            __builtin_prefetch(xin + i + stride, 0, 0);
            __builtin_prefetch(tin + i + stride, 0, 0);
        }
        float4 x = xin[i];
        float4 t = tin[i];
        float4 o;
#pragma unroll
        for (int c = 0; c < 4; ++c) {
            const float xv = (&x.x)[c];
            const float tv = (&t.x)[c];
            float pl, nl;
            bce_pair(xv, tv, pl, nl);
            float r = (tv > 0.0f) ? pl : 0.0f;
            if (tv < 0.0f && nl < thr) r += nl;   // == pos*pl + neg*mask*nl
            (&o.x)[c] = r;
        }
        out[i] = o;
    }
    int i = n4 * 4 + gid;
    if (i < n) {
        const float xv = ((const float*)xin)[i];
        const float tv = ((const float*)tin)[i];
        float pl, nl;
        bce_pair(xv, tv, pl, nl);
        float r = (tv > 0.0f) ? pl : 0.0f;
        if (tv < 0.0f && nl < thr) r += nl;
        ((float*)out)[i] = r;
    }
}

}  // namespace

// ---------------------------------------------------------------------------
extern "C" void kernel_launch(void* const* d_in, const int* in_sizes, int n_in,
                              void* d_out, int out_size, void* d_ws, size_t ws_size,
                              hipStream_t stream) {
    const float* x    = (const float*)d_in[0];
    const float* t    = (const float*)d_in[1];
    const int*   kptr = (const int*)d_in[2];

    const int n  = in_sizes[0];   // B*C = 10,240,000
    const int n4 = n / 4;

    // Workspace layout: [hist0 | hist1 | hist2 | SelState(pad to 64B) | keys?]
    unsigned* hist0 = (unsigned*)d_ws;
    unsigned* hist1 = hist0 + BINS_P01;
    unsigned* hist2 = hist1 + BINS_P01;
    SelState* st    = (SelState*)(hist2 + BINS_P2);
    unsigned* keys  = (unsigned*)((char*)st + 64);

    const size_t base_bytes = (size_t)NHIST * 4 + 64;
    const bool   cache_keys = ws_size >= base_bytes + (size_t)n * 4;

    const dim3 blk(256);
    const dim3 grd(2048);   // grid-stride; ~5 float4 per thread

    init_kernel<<<(NHIST + 255) / 256, 256, 0, stream>>>(hist0, st, kptr);

    if (cache_keys) {
        pass0_kernel<true><<<grd, blk, 0, stream>>>(
            (const float4*)x, (const float4*)t, n4, n, keys, hist0, st);
        scan_pass<0><<<1, 256, 0, stream>>>(hist0, st);
        hist_cached<1><<<grd, blk, 0, stream>>>((const uint4*)keys, n4, n, hist1, st);
        scan_pass<1><<<1, 256, 0, stream>>>(hist1, st);
        hist_cached<2><<<grd, blk, 0, stream>>>((const uint4*)keys, n4, n, hist2, st);
        scan_pass<2><<<1, 256, 0, stream>>>(hist2, st);
    } else {
        pass0_kernel<false><<<grd, blk, 0, stream>>>(
            (const float4*)x, (const float4*)t, n4, n, nullptr, hist0, st);
        scan_pass<0><<<1, 256, 0, stream>>>(hist0, st);
        hist_recompute<1><<<grd, blk, 0, stream>>>(
            (const float4*)x, (const float4*)t, n4, n, hist1, st);
        scan_pass<1><<<1, 256, 0, stream>>>(hist1, st);
        hist_recompute<2><<<grd, blk, 0, stream>>>(
            (const float4*)x, (const float4*)t, n4, n, hist2, st);
        scan_pass<2><<<1, 256, 0, stream>>>(hist2, st);
    }

    final_loss<<<grd, blk, 0, stream>>>((const float4*)x, (const float4*)t,
                                        (float4*)d_out, n4, n, st);
}